// BiRWKVLLADABlock_10222022165145
// MI455X (gfx1250) — compile-verified
//
#include <hip/hip_runtime.h>
#include <math.h>

// Problem constants (B, T, C, TD) = (4, 2048, 1024, 128)
#define B_   4
#define T_   2048
#define C_   1024
#define TD_  128
#define NC_  32          // number of T-chunks
#define L_   64          // chunk length (NC_ * L_ == T_)
#define BC_  (B_ * C_)

typedef __attribute__((ext_vector_type(2))) float v2f;
typedef __attribute__((ext_vector_type(8))) float v8f;

// ---- workspace layout (in floats) ----
#define WS_DECAY 0                        // decay[B][C]
#define WS_UCV   (WS_DECAY + BC_)         // u_c[B][C]
#define WS_TF    (WS_UCV + BC_)           // time_factor[B] (+pad)
#define WS_SFN   (WS_TF + 64)             // fwd chunk-end num   [NC][B][C]
#define WS_SFD   (WS_SFN + NC_*BC_)       // fwd chunk-end den
#define WS_SBN   (WS_SFD + NC_*BC_)       // bwd chunk-start num
#define WS_SBD   (WS_SBN + NC_*BC_)       // bwd chunk-start den
#define WS_CFN   (WS_SBD + NC_*BC_)       // fwd carry into chunk (num)
#define WS_CFD   (WS_CFN + NC_*BC_)
#define WS_CBN   (WS_CFD + NC_*BC_)       // bwd carry into chunk (num)
#define WS_CBD   (WS_CBN + NC_*BC_)

// ---------------------------------------------------------------------------
// K0: time-mixing GEMMs via V_WMMA_F32_16X16X4_F32.
//   delta_w = time_emb(4x128) @ W_time_w^T(128x1024), same for u.
//   One wave per 16-channel tile; M = batch padded to 16; K looped 4 at a time.
//   A 16x4 f32 layout: lanes 0-15: v[0]=K0, v[1]=K1 ; lanes 16-31: K2,K3.
//   B 4x16  f32 layout: lanes 0-15: v[0]=row0,v[1]=row1 ; lanes 16-31: rows 2,3.
// ---------------------------------------------------------------------------
__global__ __launch_bounds__(32)
void k0_setup(const float* __restrict__ te, const float* __restrict__ w,
              const float* __restrict__ u, const float* __restrict__ Ww,
              const float* __restrict__ Wu, const float* __restrict__ lts_p,
              float* __restrict__ ws)
{
    const int lane = threadIdx.x;
    const int c0   = blockIdx.x * 16;

    __shared__ float s_tf[B_];
    if (lane < B_) {
        float s = 0.f;
        for (int i = 0; i < TD_; ++i) s += te[lane * TD_ + i];
        s_tf[lane] = 1.f / (1.f + __expf(-s));
    }
    __syncthreads();

    const int   mrow  = lane & 15;          // M (A) / N (B) index for this lane
    const int   khalf = (lane >> 4) * 2;    // 0 for lanes 0-15, 2 for lanes 16-31
    const float amask = (mrow < B_) ? 1.f : 0.f;   // pad batch rows 4..15 with 0
    const int   arow  = (mrow < B_) ? mrow : 0;

    v8f accw = {};
    v8f accu = {};
    for (int k0 = 0; k0 < TD_; k0 += 4) {
        v2f a, bw, bu;
        a.x  = amask * te[arow * TD_ + k0 + khalf + 0];
        a.y  = amask * te[arow * TD_ + k0 + khalf + 1];
        bw.x = Ww[(c0 + mrow) * TD_ + k0 + khalf + 0];
        bw.y = Ww[(c0 + mrow) * TD_ + k0 + khalf + 1];
        bu.x = Wu[(c0 + mrow) * TD_ + k0 + khalf + 0];
        bu.y = Wu[(c0 + mrow) * TD_ + k0 + khalf + 1];
        accw = __builtin_amdgcn_wmma_f32_16x16x4_f32(
                   false, a, false, bw, (short)0, accw, false, false);
        accu = __builtin_amdgcn_wmma_f32_16x16x4_f32(
                   false, a, false, bu, (short)0, accu, false, false);
    }

    const float lts = lts_p[0];
    if (lane < 16) {                 // lanes 0-15 hold rows M=0..7 (we need 0..3)
        const int   c       = c0 + lane;
        const float wc_base = w[c];
        const float uc_base = u[c];
#pragma unroll
        for (int m = 0; m < B_; ++m) {
            const float wc  = wc_base + accw[m] * lts;
            const float dec = __expf(-__expf(wc)) * (0.5f + 0.5f * s_tf[m]);
            ws[WS_DECAY + m * C_ + c] = dec;
            ws[WS_UCV   + m * C_ + c] = uc_base + accu[m] * lts;
        }
    }
    if (blockIdx.x == 0 && lane < B_) ws[WS_TF + lane] = s_tf[lane];
}

// ---------------------------------------------------------------------------
// K1: per-chunk partials, 2 channels per thread (float2 / b64 streaming).
//   One ascending pass gives both directions:
//     fwd chunk-end:   fn = fn*d + x      (value at t1 with zero carry)
//     bwd chunk-start: bn = sum d^i * x   (value at t0 with zero carry)
// ---------------------------------------------------------------------------
__global__ __launch_bounds__(128)
void k1_chunks(const float* __restrict__ k, const float* __restrict__ v,
               float* __restrict__ ws)
{
    const int c = (blockIdx.x * 128 + threadIdx.x) * 2;   // even channel
    const int j = blockIdx.y;
    const int b = blockIdx.z;

    const float2 d = *(const float2*)&ws[WS_DECAY + b * C_ + c];

    const size_t base = ((size_t)b * T_ + (size_t)j * L_) * C_ + c;
    const float2* kp = (const float2*)(k + base);
    const float2* vp = (const float2*)(v + base);

    float2 fn = make_float2(0.f, 0.f), fd = fn, bn = fn, bd = fn;
    float2 dp = make_float2(1.f, 1.f);
    for (int i = 0; i < L_; ++i) {
        const float2 kk = kp[(size_t)i * (C_ / 2)];
        const float2 vv = vp[(size_t)i * (C_ / 2)];
        const float ekx = __expf(fminf(kk.x, 30.f));
        const float eky = __expf(fminf(kk.y, 30.f));
        const float evx = ekx * vv.x;
        const float evy = eky * vv.y;
        fn.x = fn.x * d.x + evx;  fn.y = fn.y * d.y + evy;
        fd.x = fd.x * d.x + ekx;  fd.y = fd.y * d.y + eky;
        bn.x += dp.x * evx;       bn.y += dp.y * evy;
        bd.x += dp.x * ekx;       bd.y += dp.y * eky;
        dp.x *= d.x;              dp.y *= d.y;
    }
    const int o = j * BC_ + b * C_ + c;
    *(float2*)&ws[WS_SFN + o] = fn;
    *(float2*)&ws[WS_SFD + o] = fd;
    *(float2*)&ws[WS_SBN + o] = bn;
    *(float2*)&ws[WS_SBD + o] = bd;
}

// ---------------------------------------------------------------------------
// K2: scan the 32 chunk summaries per channel -> carries entering each chunk.
// ---------------------------------------------------------------------------
__global__ __launch_bounds__(256)
void k2_scan(float* __restrict__ ws)
{
    const int bc = blockIdx.x * 256 + threadIdx.x;   // 0..B*C-1
    const float d = ws[WS_DECAY + bc];
    float dl = d;
#pragma unroll
    for (int i = 0; i < 6; ++i) dl *= dl;            // d^64

    float cn = 0.f, cd = 0.f;
    for (int j = 0; j < NC_; ++j) {
        const int o = j * BC_ + bc;
        ws[WS_CFN + o] = cn;
        ws[WS_CFD + o] = cd;
        cn = cn * dl + ws[WS_SFN + o];
        cd = cd * dl + ws[WS_SFD + o];
    }
    cn = 0.f; cd = 0.f;
    for (int j = NC_ - 1; j >= 0; --j) {
        const int o = j * BC_ + bc;
        ws[WS_CBN + o] = cn;
        ws[WS_CBD + o] = cd;
        cn = cn * dl + ws[WS_SBN + o];
        cd = cd * dl + ws[WS_SBD + o];
    }
}

// ---------------------------------------------------------------------------
// K3: per chunk — backward local scan (stash per-t values in LDS), then the
//     forward local scan fused with the epilogue and the single output write.
//     k/v re-reads hit the 192MB L2 (whole working set is resident).
//     64KB LDS @ 128 threads -> 5 workgroups (20 waves) per WGP.
// ---------------------------------------------------------------------------
__global__ __launch_bounds__(128)
void k3_final(const float* __restrict__ r, const float* __restrict__ k,
              const float* __restrict__ v, const float* __restrict__ ws,
              float* __restrict__ out)
{
    __shared__ float2 sbk[L_ * 128];                 // 64 KB per-thread scratch
    const int tid = threadIdx.x;
    const int c   = blockIdx.x * 128 + tid;
    const int j   = blockIdx.y;
    const int b   = blockIdx.z;
    const int o   = j * BC_ + b * C_ + c;

    const float d  = ws[WS_DECAY + b * C_ + c];
    const float uc = ws[WS_UCV   + b * C_ + c];
    const float tf = ws[WS_TF + b];

    const size_t base = ((size_t)b * T_ + (size_t)j * L_) * C_ + c;
    const float* kp = k + base;
    const float* vp = v + base;
    const float* rp = r + base;
    float*       op = out + base;

    // backward local scan with carry from the right
    float bn = ws[WS_CBN + o], bd = ws[WS_CBD + o];
    for (int i = L_ - 1; i >= 0; --i) {
        const float kk = kp[(size_t)i * C_];
        const float vv = vp[(size_t)i * C_];
        const float ek = __expf(fminf(kk, 30.f));
        bn = bn * d + ek * vv;
        bd = bd * d + ek;
        sbk[i * 128 + tid] = make_float2(bn, bd);
    }
    __syncthreads();

    // forward local scan + epilogue
    float fn = ws[WS_CFN + o], fd = ws[WS_CFD + o];
    const float oscale = 0.8f + 0.2f * tf;
    for (int i = 0; i < L_; ++i) {
        const float kk = kp[(size_t)i * C_];
        const float vv = vp[(size_t)i * C_];
        const float rr = rp[(size_t)i * C_];
        const float ek = __expf(fminf(kk, 30.f));
        const float ev = ek * vv;
        fn = fn * d + ev;
        fd = fd * d + ek;
        const float2 bb  = sbk[i * 128 + tid];
        const float  euk = __expf(fminf(uc + kk, 30.f));
        const float  num = fn + bb.x - ev + euk * vv;
        const float  den = fd + bb.y - ek + euk;
        const float  sig = 1.f / (1.f + __expf(-rr));
        op[(size_t)i * C_] = sig * (num / fmaxf(den, 1e-8f)) * oscale;
    }
}

// ---------------------------------------------------------------------------
extern "C" void kernel_launch(void* const* d_in, const int* in_sizes, int n_in,
                              void* d_out, int out_size, void* d_ws, size_t ws_size,
                              hipStream_t stream)
{
    const float* r   = (const float*)d_in[0];
    const float* k   = (const float*)d_in[1];
    const float* v   = (const float*)d_in[2];
    const float* te  = (const float*)d_in[3];
    const float* w   = (const float*)d_in[4];
    const float* u   = (const float*)d_in[5];
    const float* Ww  = (const float*)d_in[6];
    const float* Wu  = (const float*)d_in[7];
    const float* lts = (const float*)d_in[8];
    float* out = (float*)d_out;
    float* ws  = (float*)d_ws;

    k0_setup <<<dim3(C_ / 16),              dim3(32),  0, stream>>>(te, w, u, Ww, Wu, lts, ws);
    k1_chunks<<<dim3(C_ / 256, NC_, B_),    dim3(128), 0, stream>>>(k, v, ws);
    k2_scan  <<<dim3(BC_ / 256),            dim3(256), 0, stream>>>(ws);
    k3_final <<<dim3(C_ / 128, NC_, B_),    dim3(128), 0, stream>>>(r, k, v, ws, out);
}